// GraphConvoluation_40089224740870
// MI455X (gfx1250) — compile-verified
//
#include <hip/hip_runtime.h>

typedef __attribute__((ext_vector_type(2))) float v2f;
typedef __attribute__((ext_vector_type(8))) float v8f;

#define N_NODES 50000
#define N_EDGES 1600000
#define IN_DIM  512
#define OUT_DIM 256

// 512 + 4 pad: column reads hit distinct banks ((4*m + k) % 64), rows stay 16B-aligned
#define LDS_STRIDE 516

// Guaranteed hardware fp32 atomic (no CAS-loop expansion). No-return form
// uses STOREcnt; s_endpgm's implicit wait-idle drains it.
__device__ __forceinline__ void atomic_add_f32(float* p, float v) {
    asm volatile("global_atomic_add_f32 %0, %1, off" : : "v"(p), "v"(v) : "memory");
}

// ---------------------------------------------------------------------------
// Kernel 1: support = inputs @ weight   (fp32 WMMA 16x16x4)
// One block per 16-row M tile; 8 waves, each wave computes 2 of the 16 N-tiles.
// A tile (16x512 fp32 = 32KB) staged in LDS via async global->LDS copies.
// ---------------------------------------------------------------------------
__global__ __launch_bounds__(256) void gcn_gemm_kernel(
    const float* __restrict__ inputs, const float* __restrict__ weight,
    float* __restrict__ support)
{
    __shared__ float As[16 * LDS_STRIDE];

    const int m_base = blockIdx.x * 16;
    const int tid    = threadIdx.x;

    // Async-copy the A tile: 16 x 512 floats = 2048 x b128, 256 lanes x 8.
    // Each lane supplies its own (padded) LDS byte address in VDST.
    const float* gsrc = inputs + m_base * IN_DIM;
#pragma unroll
    for (int i = 0; i < 8; ++i) {
        int idx  = tid + i * 256;     // float4 index within tile
        int row  = idx >> 7;          // 128 float4 per row
        int col4 = idx & 127;
        unsigned lds_off = (unsigned)(uintptr_t)&As[row * LDS_STRIDE + col4 * 4];
        const float* g = gsrc + idx * 4;
        asm volatile("global_load_async_to_lds_b128 %0, %1, off"
                     : : "v"(lds_off), "v"(g) : "memory");
    }
    asm volatile("s_wait_asynccnt 0x0" : : : "memory");
    __syncthreads();

    const int wave = tid >> 5;      // 0..7
    const int lane = tid & 31;
    const int half = lane >> 4;     // 0: lanes 0-15, 1: lanes 16-31
    const int l16  = lane & 15;
    const int n0   = wave * 16;         // first N-tile column base
    const int n1   = (wave + 8) * 16;   // second N-tile column base

    v8f c0 = {};
    v8f c1 = {};
    const float* __restrict__ arow   = &As[l16 * LDS_STRIDE];  // A: M = lane&15
    const float* __restrict__ wbase0 = weight + n0 + l16;      // B: N = lane&15
    const float* __restrict__ wbase1 = weight + n1 + l16;

#pragma unroll 4
    for (int k = 0; k < IN_DIM; k += 4) {
        const int kk = k + 2 * half;   // upper half-lanes hold K+2 (ISA 7.12.2)
        v2f a;
        a.x = arow[kk];
        a.y = arow[kk + 1];
        v2f b0;
        b0.x = wbase0[kk * OUT_DIM];
        b0.y = wbase0[(kk + 1) * OUT_DIM];
        v2f b1;
        b1.x = wbase1[kk * OUT_DIM];
        b1.y = wbase1[(kk + 1) * OUT_DIM];
        // 8 args: (neg_a, A, neg_b, B, c_mod, C, reuse_a, reuse_b)
        c0 = __builtin_amdgcn_wmma_f32_16x16x4_f32(false, a, false, b0,
                                                   (short)0, c0, false, false);
        c1 = __builtin_amdgcn_wmma_f32_16x16x4_f32(false, a, false, b1,
                                                   (short)0, c1, false, false);
    }

    // C/D layout: VGPR i -> row (half*8 + i), col = lane&15
#pragma unroll
    for (int i = 0; i < 8; ++i) {
        const int row = m_base + half * 8 + i;
        support[row * OUT_DIM + n0 + l16] = c0[i];
        support[row * OUT_DIM + n1 + l16] = c1[i];
    }
}

// ---------------------------------------------------------------------------
// Kernel 2: out[n][c] = bias[c]   (float4 broadcast; also clears poison)
// ---------------------------------------------------------------------------
__global__ __launch_bounds__(256) void gcn_init_out(
    const float* __restrict__ bias, float* __restrict__ out)
{
    const int idx = blockIdx.x * 256 + threadIdx.x;   // float4 index
    const float4 b = ((const float4*)bias)[idx & (OUT_DIM / 4 - 1)];
    ((float4*)out)[idx] = b;
}

// ---------------------------------------------------------------------------
// Kernel 3: SpMM scatter. One wave per edge:
//   out[row][:] += val * support[col][:]   (256 floats, 8 per lane)
// support & out both live in L2 (51 MB each), so gathers/atomics are L2-rate.
// ---------------------------------------------------------------------------
__global__ __launch_bounds__(256) void gcn_spmm_kernel(
    const int* __restrict__ edge_row, const int* __restrict__ edge_col,
    const float* __restrict__ edge_val, const float* __restrict__ support,
    float* __restrict__ out)
{
    // Wave-uniform edge id -> SGPR, so edge metadata becomes scalar loads.
    const int e = __builtin_amdgcn_readfirstlane(
        (int)((blockIdx.x * 256 + threadIdx.x) >> 5));
    const int lane = threadIdx.x & 31;

    const int   r = edge_row[e];
    const int   c = edge_col[e];
    const float v = edge_val[e];

    const float4* __restrict__ src = (const float4*)(support + c * OUT_DIM);
    float* __restrict__ dst = out + r * OUT_DIM;

#pragma unroll
    for (int j = 0; j < 2; ++j) {
        const int f4 = lane + j * 32;       // float4 index within row
        const float4 s = src[f4];
        float* d = dst + f4 * 4;
        atomic_add_f32(d + 0, v * s.x);
        atomic_add_f32(d + 1, v * s.y);
        atomic_add_f32(d + 2, v * s.z);
        atomic_add_f32(d + 3, v * s.w);
    }
}

// ---------------------------------------------------------------------------
extern "C" void kernel_launch(void* const* d_in, const int* in_sizes, int n_in,
                              void* d_out, int out_size, void* d_ws, size_t ws_size,
                              hipStream_t stream) {
    const float* inputs   = (const float*)d_in[0];
    const int*   edge_row = (const int*)  d_in[1];
    const int*   edge_col = (const int*)  d_in[2];
    const float* edge_val = (const float*)d_in[3];
    const float* weight   = (const float*)d_in[4];
    const float* bias     = (const float*)d_in[5];

    float* out     = (float*)d_out;
    float* support = (float*)d_ws;   // 50000*256*4 = 51.2 MB scratch

    // 1) support = inputs @ weight  (3125 M-tiles, exact: 50000 = 3125*16)
    gcn_gemm_kernel<<<N_NODES / 16, 256, 0, stream>>>(inputs, weight, support);

    // 2) out = broadcast(bias)  (12.8M floats = 3.2M float4, exact /256)
    gcn_init_out<<<(N_NODES * OUT_DIM / 4) / 256, 256, 0, stream>>>(bias, out);

    // 3) scatter-accumulate edges (wave per edge; 1.6M waves = 200000 blocks)
    gcn_spmm_kernel<<<N_EDGES / 8, 256, 0, stream>>>(edge_row, edge_col,
                                                     edge_val, support, out);
}